// KNNGraph_67997922230585
// MI455X (gfx1250) — compile-verified
//
#include <hip/hip_runtime.h>

// KNN graph (per-batch 32-NN) for MI455X / gfx1250.
// One wave32 owns 32 queries. Per 16-ref tile: two V_WMMA_F32_16X16X4_F32
// (queries 0-15 and 16-31 vs 16 refs, xyz padded to 4 with 0) produce all
// 32x16 dot products. Lanes deposit (|r|^2 - 2*dot) transposed into a
// bank-conflict-free LDS tile; then each lane owns exactly one query and does
// threshold-filtered sorted insertion into its lane-private top-32 list in
// LDS (threshold cached in a register). No cross-lane serialization at all.
// Batch ranges from an atomic pre-pass restrict the scan; the per-pair
// batch-equality mask reproduces the reference's +inf masking exactly.

#define KNN_K 32
#define NB_MAX 64
#define QPW 32          // queries per wave
#define CSTRIDE 17      // padded tile stride (17 coprime with 64 banks)
#define LSTRIDE 33      // padded list stride (33 coprime with 64 banks)

typedef __attribute__((ext_vector_type(2))) float v2f;
typedef __attribute__((ext_vector_type(4))) float v4f;
typedef __attribute__((ext_vector_type(8))) float v8f;

__global__ void knn_init_ranges(int* bstart, int* bend, int N) {
    int i = threadIdx.x;
    if (i < NB_MAX) { bstart[i] = N; bend[i] = 0; }
}

__global__ void knn_scan_ranges(const float* __restrict__ ref,
                                int* bstart, int* bend, int N) {
    int i = blockIdx.x * blockDim.x + threadIdx.x;
    if (i >= N) return;
    int b = (int)ref[(size_t)i * 4];
    b = min(max(b, 0), NB_MAX - 1);
    atomicMin(&bstart[b], i);
    atomicMax(&bend[b], i + 1);
}

__global__ __launch_bounds__(32)
void knn_topk_kernel(const float* __restrict__ ref,
                     const float* __restrict__ qry,
                     const int* __restrict__ bstart,
                     const int* __restrict__ bend,
                     float* __restrict__ out,
                     int N, int M) {
    __shared__ float s_c[QPW * CSTRIDE];       // rn - 2*dot, owner-major
    __shared__ int   s_rb[16];                 // ref batch ids for the tile
    __shared__ float s_dist[QPW * LSTRIDE];    // lane-private sorted lists
    __shared__ int   s_idx [QPW * LSTRIDE];

    const int lane  = threadIdx.x;
    const int half  = lane >> 4;       // 0: lanes 0-15 (K=0,1), 1: lanes 16-31 (K=2,3)
    const int ln    = lane & 15;
    const int qbase = blockIdx.x * QPW;

    // ---- per-lane owned query (selection state lives with this lane) ----
    int qg = qbase + lane;
    v4f q = {0.f, 0.f, 0.f, 0.f};
    int qb = -1;
    if (qg < M) { q = *(const v4f*)(qry + (size_t)qg * 4); qb = (int)q.x; }
    float qn = q.y * q.y + q.z * q.z + q.w * q.w;
    float qz = q.w;

    float* dl = &s_dist[lane * LSTRIDE];
    int*   il = &s_idx [lane * LSTRIDE];
#pragma unroll
    for (int j = 0; j < KNN_K; ++j) { dl[j] = __builtin_inff(); il[j] = 0; }
    float thr = __builtin_inff();

    // ---- A matrices (16x4 f32): tile0 = queries qbase+0..15, tile1 = +16..31
    // lanes 0-15 hold K=0,1 = (x,y); lanes 16-31 hold K=2,3 = (z,0)
    int qr0 = qbase + ln, qr1 = qbase + 16 + ln;
    v4f qa0 = {0.f,0.f,0.f,0.f}, qa1 = {0.f,0.f,0.f,0.f};
    if (qr0 < M) qa0 = *(const v4f*)(qry + (size_t)qr0 * 4);
    if (qr1 < M) qa1 = *(const v4f*)(qry + (size_t)qr1 * 4);
    v2f a0, a1;
    a0.x = half ? qa0.w : qa0.y;  a0.y = half ? 0.0f : qa0.z;
    a1.x = half ? qa1.w : qa1.y;  a1.y = half ? 0.0f : qa1.z;

    // ---- uniform ref range for this wave (queries are batch-sorted) ----
    int qb_lo = __shfl(qb, 0);
    int qb_hi = __shfl(qb, QPW - 1);
    if (qb_lo < 0) qb_lo = 0;
    if (qb_hi < qb_lo) qb_hi = qb_lo;
    if (qb_hi > NB_MAX - 1) qb_hi = NB_MAX - 1;
    int lo = N, hi = 0;
    for (int b = qb_lo; b <= qb_hi; ++b) {
        lo = min(lo, bstart[b]);
        hi = max(hi, bend[b]);
    }
    lo &= ~15;

    for (int r0 = lo; r0 < hi; r0 += 16) {
        __builtin_prefetch(ref + (size_t)(r0 + 16) * 4, 0, 3);

        // each lane loads ref column n = ln (both halves load the same ref)
        int rI = r0 + ln;
        v4f rv = {0.f, 0.f, 0.f, 0.f};
        int rb = -2;
        if (rI < N) { rv = *(const v4f*)(ref + (size_t)rI * 4); rb = (int)rv.x; }
        float rn = rv.y * rv.y + rv.z * rv.z + rv.w * rv.w;
        if (half == 0) s_rb[ln] = rb;

        // B matrix (4x16 f32): lanes 0-15 = (x,y); lanes 16-31 = (z,0)
        v2f bm;
        bm.x = half ? rv.w : rv.y;
        bm.y = half ? 0.0f : rv.z;

        v8f c0 = {}, c1 = {};
        c0 = __builtin_amdgcn_wmma_f32_16x16x4_f32(false, a0, false, bm,
                                                   (short)0, c0, false, false);
        c1 = __builtin_amdgcn_wmma_f32_16x16x4_f32(false, a1, false, bm,
                                                   (short)0, c1, false, false);

        // transpose to owner-major: s_c[m*CSTRIDE + n] = rn - 2*dot(q_m, r_n)
        // C layout: lane column n = ln, rows m = 8*half + v
#pragma unroll
        for (int v = 0; v < 8; ++v) {
            int m = half * 8 + v;
            s_c[m * CSTRIDE + ln]        = rn - 2.0f * c0[v];
            s_c[(16 + m) * CSTRIDE + ln] = rn - 2.0f * c1[v];
        }
        __syncthreads();

        // lane-private selection: 16 candidates for this lane's query
#pragma unroll 4
        for (int n = 0; n < 16; ++n) {
            float d2 = qn + s_c[lane * CSTRIDE + n];
            if (s_rb[n] == qb && d2 < thr) {
                int j = KNN_K - 1;
                while (j > 0 && dl[j - 1] > d2) {
                    dl[j] = dl[j - 1];
                    il[j] = il[j - 1];
                    --j;
                }
                dl[j] = d2;
                il[j] = r0 + n;
                thr = dl[KNN_K - 1];
            }
        }
        __syncthreads();   // protect s_c/s_rb before next tile overwrites
    }

    // ---- emit (e_ref, e_query, mask) as concatenated float arrays ----
    if (qg < M) {
        size_t MK = (size_t)M * KNN_K;
#pragma unroll 4
        for (int k = 0; k < KNN_K; ++k) {
            int rid = il[k];
            size_t o = (size_t)qg * KNN_K + k;
            out[o]      = (float)rid;
            out[MK + o] = (float)qg;
            float rz = ref[(size_t)rid * 4 + 3];
            out[2 * MK + o] = ((qz - rz) >= -1e-5f) ? 1.0f : 0.0f;
        }
    }
}

extern "C" void kernel_launch(void* const* d_in, const int* in_sizes, int n_in,
                              void* d_out, int out_size, void* d_ws, size_t ws_size,
                              hipStream_t stream) {
    const float* ref = (const float*)d_in[0];   // ref_bxyz   (N,4)
    const float* qry = (const float*)d_in[1];   // query_bxyz (M,4)
    int N = in_sizes[0] / 4;
    int M = in_sizes[1] / 4;

    int* bstart = (int*)d_ws;
    int* bend   = bstart + NB_MAX;

    hipLaunchKernelGGL(knn_init_ranges, dim3(1), dim3(NB_MAX), 0, stream,
                       bstart, bend, N);
    hipLaunchKernelGGL(knn_scan_ranges, dim3((N + 255) / 256), dim3(256), 0, stream,
                       ref, bstart, bend, N);
    hipLaunchKernelGGL(knn_topk_kernel, dim3((M + QPW - 1) / QPW), dim3(32), 0, stream,
                       ref, qry, bstart, bend, (float*)d_out, N, M);
}